// NMN_18073222382300
// MI455X (gfx1250) — compile-verified
//
#include <hip/hip_runtime.h>
#include <math.h>

typedef __attribute__((ext_vector_type(2))) float v2f;
typedef __attribute__((ext_vector_type(8))) float v8f;

#define N_   64
#define HW_  784
#define D_   512
#define L_   8
#define M_   8

__device__ __forceinline__ float wave_sum(float v) {
#pragma unroll
    for (int o = 16; o > 0; o >>= 1) v += __shfl_xor(v, o, 32);
    return v;
}

// Out[Mrows x Nc] = X[Mrows x K] @ W[K x Nc] + bias  (row-major, fp32)
// One wave per 16x16 output tile, V_WMMA_F32_16X16X4_F32 along K.
// A frag (16x4): lanes0-15 K={0,1}, lanes16-31 K={2,3}, M = lane&15.
// B frag (4x16): lanes0-15 rows K={0,1}, lanes16-31 rows K={2,3}, N = lane&15.
// C/D (16x16): vgpr r -> row r (lanes0-15) / r+8 (lanes16-31), col = lane&15.
__global__ void gemm_wmma_f32(const float* __restrict__ X,
                              const float* __restrict__ W,
                              const float* __restrict__ bias,
                              float* __restrict__ Out,
                              int Mrows, int K, int Nc) {
    const int wave = (blockIdx.x * blockDim.x + threadIdx.x) >> 5;
    const int lane = threadIdx.x & 31;
    const int tilesN = Nc >> 4;
    const int totalTiles = (Mrows >> 4) * tilesN;
    if (wave >= totalTiles) return;               // wave-uniform exit (EXEC stays full)
    const int tm   = (wave / tilesN) << 4;
    const int tn   = (wave % tilesN) << 4;
    const int half = lane >> 4;                   // 0 or 1
    const int l16  = lane & 15;
    const int m    = tm + l16;
    const int nn   = tn + l16;
    const int kh   = half << 1;                   // 0 or 2
    v8f acc = {};
    for (int k = 0; k < K; k += 4) {
        v2f a, b;
        a.x = X[m * K + k + kh];
        a.y = X[m * K + k + kh + 1];
        b.x = W[(size_t)(k + kh)     * Nc + nn];
        b.y = W[(size_t)(k + kh + 1) * Nc + nn];
        acc = __builtin_amdgcn_wmma_f32_16x16x4_f32(false, a, false, b,
                                                    (short)0, acc, false, false);
    }
    const float bv = bias[nn];
    const int rowOff = half << 3;
#pragma unroll
    for (int r = 0; r < 8; ++r)
        Out[(size_t)(tm + rowOff + r) * Nc + nn] = acc[r] + bv;
}

// att_in = read(att_stack_prev, stack_ptr_prev); spatial softmax -> w_att[n,784]
__global__ void att_softmax_kernel(const float* __restrict__ stk,
                                   const float* __restrict__ ptr,
                                   float* __restrict__ w_att) {
    const int n = blockIdx.x, tid = threadIdx.x;
    __shared__ float s_att[HW_];
    __shared__ float sred[256];
    float p[L_];
#pragma unroll
    for (int l = 0; l < L_; ++l) p[l] = ptr[n * L_ + l];
    float lmax = -3.4e38f;
    for (int hw = tid; hw < HW_; hw += 256) {
        const float4* s4 = reinterpret_cast<const float4*>(stk + ((size_t)(n * HW_ + hw)) * L_);
        float4 a = s4[0], b = s4[1];
        float v = a.x*p[0] + a.y*p[1] + a.z*p[2] + a.w*p[3]
                + b.x*p[4] + b.y*p[5] + b.z*p[6] + b.w*p[7];
        s_att[hw] = v;
        lmax = fmaxf(lmax, v);
    }
    sred[tid] = lmax; __syncthreads();
    for (int s = 128; s > 0; s >>= 1) { if (tid < s) sred[tid] = fmaxf(sred[tid], sred[tid + s]); __syncthreads(); }
    const float mx = sred[0]; __syncthreads();
    float lsum = 0.f;
    for (int hw = tid; hw < HW_; hw += 256) {
        float e = expf(s_att[hw] - mx);
        s_att[hw] = e; lsum += e;
    }
    sred[tid] = lsum; __syncthreads();
    for (int s = 128; s > 0; s >>= 1) { if (tid < s) sred[tid] += sred[tid + s]; __syncthreads(); }
    const float inv = 1.0f / sred[0];
    for (int hw = tid; hw < HW_; hw += 256) w_att[n * HW_ + hw] = s_att[hw] * inv;
}

// kb_att[n,d] = sum_hw kb[n,hw,d] * w_att[n,hw]
// 128 threads per n, each owning 4 consecutive d's -> b128 loads/stores.
__global__ void kb_att_kernel(const float* __restrict__ kb,
                              const float* __restrict__ w_att,
                              float* __restrict__ katt) {
    const int n = blockIdx.x;
    const int t = threadIdx.x;                           // 0..127
    const float4* kbn = reinterpret_cast<const float4*>(kb + (size_t)n * HW_ * D_);
    const float* wn = w_att + n * HW_;
    float4 acc = make_float4(0.f, 0.f, 0.f, 0.f);
    for (int hw = 0; hw < HW_; ++hw) {
        if (hw + 8 < HW_)
            __builtin_prefetch(&kbn[(size_t)(hw + 8) * (D_ / 4) + t], 0, 3);
        const float w = wn[hw];
        const float4 kv = kbn[(size_t)hw * (D_ / 4) + t];
        acc.x = fmaf(kv.x, w, acc.x);
        acc.y = fmaf(kv.y, w, acc.y);
        acc.z = fmaf(kv.z, w, acc.z);
        acc.w = fmaf(kv.w, w, acc.w);
    }
    reinterpret_cast<float4*>(katt + n * D_)[t] = acc;
}

// ep = l2norm(cd * kb_att); v = c2 * kb_att; concat = [control | mem_prev | ep]
__global__ void prep_kernel(const float* __restrict__ c2,
                            const float* __restrict__ cd,
                            const float* __restrict__ katt,
                            const float* __restrict__ control,
                            const float* __restrict__ mem_prev,
                            float* __restrict__ v,
                            float* __restrict__ concat) {
    const int n = blockIdx.x, tid = threadIdx.x;
    __shared__ float sred[256];
    const int d0 = tid, d1 = tid + 256;
    const float k0 = katt[n * D_ + d0], k1 = katt[n * D_ + d1];
    const float t0 = cd[n * D_ + d0] * k0;
    const float t1 = cd[n * D_ + d1] * k1;
    sred[tid] = t0 * t0 + t1 * t1; __syncthreads();
    for (int s = 128; s > 0; s >>= 1) { if (tid < s) sred[tid] += sred[tid + s]; __syncthreads(); }
    const float inv = 1.0f / fmaxf(sqrtf(sred[0]), 1e-12f);
    v[n * D_ + d0] = c2[n * D_ + d0] * k0;
    v[n * D_ + d1] = c2[n * D_ + d1] * k1;
    float* cc = concat + (size_t)n * (3 * D_);
    cc[d0] = control[n * D_ + d0];       cc[d1] = control[n * D_ + d1];
    cc[D_ + d0] = mem_prev[n * D_ + d0]; cc[D_ + d1] = mem_prev[n * D_ + d1];
    cc[2 * D_ + d0] = t0 * inv;          cc[2 * D_ + d1] = t1 * inv;
}

// Fused per-pixel pass: Find / Transform / Scene attention maps.
// One wave per pixel; each lane streams float4 (global_load_b128) chunks of D.
__global__ void pixel_kernel(const float* __restrict__ kb,
                             const float* __restrict__ c1,
                             const float* __restrict__ v,
                             const float* __restrict__ wf,
                             const float* __restrict__ wt,
                             const float* __restrict__ wsc,
                             const float* __restrict__ bf,
                             const float* __restrict__ bt,
                             const float* __restrict__ bs,
                             float* __restrict__ attF,
                             float* __restrict__ attT,
                             float* __restrict__ attS) {
    const int wave = (blockIdx.x * blockDim.x + threadIdx.x) >> 5;
    const int lane = threadIdx.x & 31;
    const int n  = wave / HW_;
    const int hw = wave % HW_;
    const float4* kb4 = reinterpret_cast<const float4*>(kb + ((size_t)(n * HW_ + hw)) * D_);
    const float4* c14 = reinterpret_cast<const float4*>(c1 + n * D_);
    const float4* v4  = reinterpret_cast<const float4*>(v + n * D_);
    const float4* wf4 = reinterpret_cast<const float4*>(wf);
    const float4* wt4 = reinterpret_cast<const float4*>(wt);
    const float4* ws4 = reinterpret_cast<const float4*>(wsc);
    float dot1 = 0, sq1 = 0, dot2 = 0, sq2 = 0, dot3 = 0, sq3 = 0;
#pragma unroll
    for (int it = 0; it < D_ / 128; ++it) {       // 4 iterations of 32 float4s
        const int i = it * 32 + lane;
        const float4 kv = kb4[i];
        const float4 cv = c14[i];
        const float4 vv = v4[i];
        const float4 fv = wf4[i];
        const float4 tv = wt4[i];
        const float4 sv = ws4[i];
#define ACC_COMP(c)                                                         \
        { const float kc = kv.c;                                            \
          const float t1 = kc * cv.c;                                       \
          const float t2 = kc * vv.c;                                       \
          dot1 = fmaf(t1, fv.c, dot1);  sq1 = fmaf(t1, t1, sq1);            \
          dot2 = fmaf(t2, tv.c, dot2);  sq2 = fmaf(t2, t2, sq2);            \
          dot3 = fmaf(kc, sv.c, dot3);  sq3 = fmaf(kc, kc, sq3); }
        ACC_COMP(x) ACC_COMP(y) ACC_COMP(z) ACC_COMP(w)
#undef ACC_COMP
    }
    dot1 = wave_sum(dot1); sq1 = wave_sum(sq1);
    dot2 = wave_sum(dot2); sq2 = wave_sum(sq2);
    dot3 = wave_sum(dot3); sq3 = wave_sum(sq3);
    if (lane == 0) {
        const int idx = n * HW_ + hw;
        attF[idx] = dot1 / fmaxf(sqrtf(sq1), 1e-12f) + bf[0];
        attT[idx] = dot2 / fmaxf(sqrtf(sq2), 1e-12f) + bt[0];
        attS[idx] = dot3 / fmaxf(sqrtf(sq3), 1e-12f) + bs[0];
    }
}

// Mix the 8 module stacks per element
__global__ void combine_att_kernel(const float* __restrict__ stk,
                                   const float* __restrict__ ptr,
                                   const float* __restrict__ prob,
                                   const float* __restrict__ attFb,
                                   const float* __restrict__ attTb,
                                   const float* __restrict__ attSb,
                                   float* __restrict__ out) {
    const int n = blockIdx.x, tid = threadIdx.x;
    float p[L_], pfw[L_], pbw[L_], p3[L_], pr[M_];
#pragma unroll
    for (int l = 0; l < L_; ++l) p[l] = ptr[n * L_ + l];
#pragma unroll
    for (int m = 0; m < M_; ++m) pr[m] = prob[n * M_ + m];
#pragma unroll
    for (int l = 0; l < L_; ++l) {
        pfw[l] = (l > 0)      ? p[l - 1] : 0.f;   // move_fw
        pbw[l] = (l < L_ - 1) ? p[l + 1] : 0.f;   // move_bw
        p3[l]  = (l < L_ - 1) ? p[l]     : 0.f;   // move_bw(move_fw(p))
    }
    for (int hw = tid; hw < HW_; hw += 256) {
        const size_t base = ((size_t)(n * HW_ + hw)) * L_;
        const float4* s4 = reinterpret_cast<const float4*>(stk + base);
        float4 a = s4[0], b = s4[1];
        float s[8] = {a.x, a.y, a.z, a.w, b.x, b.y, b.z, b.w};
        const int idx = n * HW_ + hw;
        const float aF = attFb[idx], aT = attTb[idx], aS = attSb[idx];
        float r_p = 0.f, r_bw = 0.f, a2f = 0.f, a1f = 0.f;
#pragma unroll
        for (int l = 0; l < 8; ++l) {
            r_p  += s[l] * p[l];
            r_bw += s[l] * pbw[l];
            const float sf = aF * pfw[l] + s[l] * (1.f - pfw[l]);  // Find stack
            a2f += sf * pfw[l];
            a1f += sf * p3[l];
        }
        const float mn  = fminf(r_bw, r_p);     // And
        const float mx  = fmaxf(r_bw, r_p);     // Or
        const float mnf = fminf(a1f, a2f);      // Filter
        float o[8];
#pragma unroll
        for (int l = 0; l < 8; ++l) {
            const float sf   = aF * pfw[l] + s[l] * (1.f - pfw[l]);
            const float oT   = aT * p[l]   + s[l] * (1.f - p[l]);
            const float oFil = mnf * p3[l] + sf   * (1.f - p3[l]);
            const float oAnd = mn * pbw[l] + s[l] * (1.f - pbw[l]);
            const float oOr  = mx * pbw[l] + s[l] * (1.f - pbw[l]);
            const float oSc  = aS * pfw[l] + s[l] * (1.f - pfw[l]);
            const float oD1  = s[l] * (1.f - p[l]);
            o[l] = pr[0]*s[l] + pr[1]*sf + pr[2]*oT + pr[3]*oFil
                 + pr[4]*oAnd + pr[5]*oOr + pr[6]*oSc + pr[7]*oD1;
        }
        float4* o4 = reinterpret_cast<float4*>(out + base);
        o4[0] = make_float4(o[0], o[1], o[2], o[3]);
        o4[1] = make_float4(o[4], o[5], o[6], o[7]);
    }
}

__global__ void ptr_kernel(const float* __restrict__ ptr,
                           const float* __restrict__ prob,
                           float* __restrict__ out) {
    const int n = blockIdx.x * blockDim.x + threadIdx.x;
    if (n >= N_) return;
    float p[L_], pr[M_], pm[L_];
#pragma unroll
    for (int l = 0; l < L_; ++l) p[l] = ptr[n * L_ + l];
#pragma unroll
    for (int m = 0; m < M_; ++m) pr[m] = prob[n * M_ + m];
    const float wp  = pr[0] + pr[2] + pr[7];
    const float wfw = pr[1] + pr[6];
    const float wbw = pr[4] + pr[5];
    float mx = -3.4e38f;
#pragma unroll
    for (int l = 0; l < L_; ++l) {
        pm[l] = wp * p[l]
              + wfw * ((l > 0) ? p[l - 1] : 0.f)
              + wbw * ((l < L_ - 1) ? p[l + 1] : 0.f)
              + pr[3] * ((l < L_ - 1) ? p[l] : 0.f);
        mx = fmaxf(mx, pm[l]);
    }
    float sum = 0.f;
#pragma unroll
    for (int l = 0; l < L_; ++l) { pm[l] = expf(pm[l] - mx); sum += pm[l]; }
    const float inv = 1.0f / sum;
#pragma unroll
    for (int l = 0; l < L_; ++l) out[n * L_ + l] = pm[l] * inv;
}

__global__ void mem_kernel(const float* __restrict__ prob,
                           const float* __restrict__ mem_prev,
                           const float* __restrict__ mem_out,
                           float* __restrict__ out) {
    const int i = blockIdx.x * blockDim.x + threadIdx.x;   // over N_*D_
    if (i >= N_ * D_) return;
    const int n = i >> 9;
    out[i] = prob[n * M_ + 0] * mem_prev[i] + prob[n * M_ + 7] * mem_out[i];
}

extern "C" void kernel_launch(void* const* d_in, const int* in_sizes, int n_in,
                              void* d_out, int out_size, void* d_ws, size_t ws_size,
                              hipStream_t stream) {
    (void)in_sizes; (void)n_in; (void)out_size; (void)ws_size;
    const float* control  = (const float*)d_in[0];
    const float* kb       = (const float*)d_in[1];
    const float* prob     = (const float*)d_in[2];
    const float* mem_prev = (const float*)d_in[3];
    const float* stk      = (const float*)d_in[4];
    const float* sptr     = (const float*)d_in[5];
    const float* find_w   = (const float*)d_in[6];
    const float* find_b   = (const float*)d_in[7];
    const float* find_cw  = (const float*)d_in[8];
    const float* find_cb  = (const float*)d_in[9];
    const float* tr_w     = (const float*)d_in[10];
    const float* tr_b     = (const float*)d_in[11];
    const float* tr_cw    = (const float*)d_in[12];
    const float* tr_cb    = (const float*)d_in[13];
    const float* sc_cw    = (const float*)d_in[14];
    const float* sc_cb    = (const float*)d_in[15];
    const float* d1_w     = (const float*)d_in[16];
    const float* d1_b     = (const float*)d_in[17];
    const float* d1_mw    = (const float*)d_in[18];
    const float* d1_mb    = (const float*)d_in[19];

    float* ws     = (float*)d_ws;
    float* c1     = ws;                 // 64*512
    float* c2     = c1 + N_ * D_;
    float* cd     = c2 + N_ * D_;
    float* w_att  = cd + N_ * D_;       // 64*784
    float* katt   = w_att + N_ * HW_;   // 64*512
    float* vvec   = katt + N_ * D_;     // 64*512
    float* concat = vvec + N_ * D_;     // 64*1536
    float* memout = concat + N_ * 3 * D_;
    float* attF   = memout + N_ * D_;   // 64*784 x3
    float* attT   = attF + N_ * HW_;
    float* attS   = attT + N_ * HW_;

    float* out_att = (float*)d_out;                 // N*H*W*L
    float* out_ptr = out_att + N_ * HW_ * L_;       // N*L
    float* out_mem = out_ptr + N_ * L_;             // N*MEM

    // 1) control projections: three 64x512x512 fp32 WMMA GEMMs
    gemm_wmma_f32<<<16, 256, 0, stream>>>(control, find_w, find_b, c1, 64, 512, 512);
    gemm_wmma_f32<<<16, 256, 0, stream>>>(control, tr_w,   tr_b,   c2, 64, 512, 512);
    gemm_wmma_f32<<<16, 256, 0, stream>>>(control, d1_w,   d1_b,   cd, 64, 512, 512);
    // 2) shared stack-read + spatial softmax (Transform & DescribeOne)
    att_softmax_kernel<<<N_, 256, 0, stream>>>(stk, sptr, w_att);
    // 3) kb pass #1: attention pooling (b128 streaming + prefetch)
    kb_att_kernel<<<N_, 128, 0, stream>>>(kb, w_att, katt);
    // 4) ep/v/concat prep
    prep_kernel<<<N_, 256, 0, stream>>>(c2, cd, katt, control, mem_prev, vvec, concat);
    // 5) DescribeOne memory GEMM: 64x1536x512 fp32 WMMA
    gemm_wmma_f32<<<16, 256, 0, stream>>>(concat, d1_mw, d1_mb, memout, 64, 1536, 512);
    // 6) kb pass #2: fused Find/Transform/Scene per-pixel reductions (b128 streaming)
    pixel_kernel<<<(N_ * HW_) / 8, 256, 0, stream>>>(kb, c1, vvec, find_cw, tr_cw, sc_cw,
                                                     find_cb, tr_cb, sc_cb, attF, attT, attS);
    // 7) module-probability mixing
    combine_att_kernel<<<N_, 256, 0, stream>>>(stk, sptr, prob, attF, attT, attS, out_att);
    ptr_kernel<<<1, 64, 0, stream>>>(sptr, prob, out_ptr);
    mem_kernel<<<(N_ * D_ + 255) / 256, 256, 0, stream>>>(prob, mem_prev, memout, out_mem);
}